// APPNP_Layer_76467597738460
// MI455X (gfx1250) — compile-verified
//
#include <hip/hip_runtime.h>

typedef __attribute__((ext_vector_type(2))) float v2f;
typedef __attribute__((ext_vector_type(8))) float v8f;

constexpr int kNodes = 50000;
constexpr int kEdges = 800000;
constexpr int kInF   = 512;
constexpr int kOutF  = 96;             // 6 tiles of 16
constexpr float kAlpha = 0.1f;
constexpr int kIters = 10;
constexpr int kColTiles = kOutF / 16;  // 6
constexpr int kRowTiles = kNodes / 16; // 3125 (exact)
constexpr int kWavesPerBlock = 4;
constexpr int kKChunk = 128;                    // K staged per LDS round
constexpr int kLdsPairs = (kKChunk / 2) * kOutF; // 6144 float2 = 48 KB

// -----------------------------------------------------------------------------
// GEMM: support = X @ W via V_WMMA_F32_16X16X4_F32 (fp32 matrix pipe).
// W is staged through LDS in 48 KB K-chunks, pre-paired so each lane's B
// fragment (W[k][col], W[k+1][col]) is a single ds_load_b64 — this keeps the
// vmem pipe free for streaming X (102 MB, the HBM-roofline term at 23.3 TB/s).
// Each wave owns one 16-row strip and all 6 column tiles (48 acc VGPRs).
// Also seeds h = support (iteration state in d_out) and acc = ALPHA*support.
// -----------------------------------------------------------------------------
__global__ __launch_bounds__(128) void appnp_gemm_wmma(
    const float* __restrict__ X, const float* __restrict__ W,
    float* __restrict__ support, float* __restrict__ h,
    float* __restrict__ acc)
{
  __shared__ v2f ldsW[kLdsPairs];  // [pair m][col]: (W[2m][col], W[2m+1][col])

  const int wave = threadIdx.x >> 5;
  const int lane = threadIdx.x & 31;
  const int half = lane >> 4;   // 0: lanes 0-15, 1: lanes 16-31
  const int lr   = lane & 15;

  const int rowTile = blockIdx.x * kWavesPerBlock + wave; // wave-uniform
  const bool active = rowTile < kRowTiles;                // uniform per wave
  const int row0 = rowTile * 16;

  v8f c[kColTiles] = {};  // zero accumulators

  const float* __restrict__ xrow = X + (size_t)(row0 + lr) * kInF; // no deref if !active

  for (int kc = 0; kc < kInF; kc += kKChunk) {
    __syncthreads();  // previous chunk fully consumed before overwrite
    // Cooperative stage: pair index m covers K rows (kc+2m, kc+2m+1)
    for (int i = threadIdx.x; i < kLdsPairs; i += blockDim.x) {
      const int m   = i / kOutF;
      const int col = i - m * kOutF;
      const int k0  = kc + 2 * m;
      v2f p;
      p.x = W[(size_t)k0 * kOutF + col];
      p.y = W[(size_t)(k0 + 1) * kOutF + col];
      ldsW[i] = p;
    }
    __syncthreads();

    if (active) {  // wave-uniform -> EXEC all-1s inside (WMMA requirement)
      for (int kb = 0; kb < kKChunk; kb += 4) {
        // A 16x4 fp32 layout: VGPR0 = K(2*half), VGPR1 = K(2*half+1), M = lr
        v2f a;
        a.x = xrow[kc + kb + 2 * half + 0];
        a.y = xrow[kc + kb + 2 * half + 1];
        const int mIdx = (kb >> 1) + half;  // pair row for this lane-half
#pragma unroll
        for (int t = 0; t < kColTiles; ++t) {
          const v2f b = ldsW[mIdx * kOutF + t * 16 + lr];  // single ds_load_b64
          c[t] = __builtin_amdgcn_wmma_f32_16x16x4_f32(
              /*neg_a=*/false, a, /*neg_b=*/false, b,
              /*c_mod=*/(short)0, c[t], /*reuse_a=*/false, /*reuse_b=*/false);
        }
      }
    }
  }

  if (active) {
    // D layout: VGPR i -> M = i + 8*half, N = lr (within the 16-col tile)
#pragma unroll
    for (int t = 0; t < kColTiles; ++t) {
#pragma unroll
      for (int i = 0; i < 8; ++i) {
        const int r = row0 + i + 8 * half;
        const size_t o = (size_t)r * kOutF + t * 16 + lr;
        const float v = c[t][i];
        support[o] = v;
        h[o]       = v;              // iteration state s_0 = support
        acc[o]     = kAlpha * v;     // seed for iteration 1 scatter
      }
    }
  }
}

// -----------------------------------------------------------------------------
// Edge scatter: one thread per (edge, float4 chunk of 4 features).
// acc[dst, 4f..] += 0.9 * val * h[src, 4f..]
// h gather is a global_load_b128 (L2-resident: h is 19.2 MB vs 192 MB L2);
// scatter via 4x global_atomic_add_f32 resolving in L2.
// -----------------------------------------------------------------------------
constexpr int kChunks = kOutF / 4;  // 24
__global__ __launch_bounds__(256) void appnp_scatter(
    const int* __restrict__ esrc, const int* __restrict__ edst,
    const float* __restrict__ eval,
    const float* __restrict__ h, float* __restrict__ acc)
{
  const unsigned tid = blockIdx.x * blockDim.x + threadIdx.x;
  const unsigned total = (unsigned)kEdges * (unsigned)kChunks;
  if (tid >= total) return;
  const unsigned e = tid / kChunks;
  const unsigned cidx = tid - e * kChunks;

  const int s = esrc[e];
  const int d = edst[e];
  const float w = (1.0f - kAlpha) * eval[e];

  const float4 hv = reinterpret_cast<const float4*>(h)[(size_t)s * kChunks + cidx];
  float* __restrict__ base = &acc[(size_t)d * kOutF + cidx * 4];
  atomicAdd(base + 0, w * hv.x);
  atomicAdd(base + 1, w * hv.y);
  atomicAdd(base + 2, w * hv.z);
  atomicAdd(base + 3, w * hv.w);
}

// -----------------------------------------------------------------------------
// Combine: h = relu(acc); acc = ALPHA*support (re-seed for next iteration).
// -----------------------------------------------------------------------------
__global__ __launch_bounds__(256) void appnp_combine(
    const float* __restrict__ support, float* __restrict__ acc,
    float* __restrict__ h)
{
  const unsigned i = blockIdx.x * blockDim.x + threadIdx.x;
  if (i >= (unsigned)kNodes * (unsigned)kOutF) return;
  h[i]   = fmaxf(acc[i], 0.0f);
  acc[i] = kAlpha * support[i];
}

extern "C" void kernel_launch(void* const* d_in, const int* in_sizes, int n_in,
                              void* d_out, int out_size, void* d_ws, size_t ws_size,
                              hipStream_t stream) {
  const float* x    = (const float*)d_in[0];   // [N, 512]
  const float* w    = (const float*)d_in[1];   // [512, 96]
  const int*   esrc = (const int*)d_in[2];     // [E]
  const int*   edst = (const int*)d_in[3];     // [E]
  const float* ev   = (const float*)d_in[4];   // [E]

  float* h       = (float*)d_out;              // iteration state, final answer
  float* support = (float*)d_ws;               // [N*96]
  float* acc     = support + (size_t)kNodes * kOutF;  // [N*96]

  // 1) fp32 WMMA GEMM (B via LDS) + seed state buffers
  const int gemmBlocks = (kRowTiles + kWavesPerBlock - 1) / kWavesPerBlock;
  appnp_gemm_wmma<<<gemmBlocks, 128, 0, stream>>>(x, w, support, h, acc);

  // 2) 10 propagation rounds
  const unsigned scatterThreads = (unsigned)kEdges * (unsigned)kChunks;
  const int scatterBlocks = (int)((scatterThreads + 255u) / 256u);
  const unsigned nodeElems = (unsigned)kNodes * (unsigned)kOutF;
  const int combineBlocks = (int)((nodeElems + 255u) / 256u);

  for (int it = 0; it < kIters; ++it) {
    appnp_scatter<<<scatterBlocks, 256, 0, stream>>>(esrc, edst, ev, h, acc);
    appnp_combine<<<combineBlocks, 256, 0, stream>>>(support, acc, h);
  }
}